// RelationAwareAttention_63170378989850
// MI455X (gfx1250) — compile-verified
//
#include <hip/hip_runtime.h>
#include <hip/hip_bf16.h>

typedef __attribute__((ext_vector_type(2))) float v2f;
typedef __attribute__((ext_vector_type(4))) float v4f;
typedef __attribute__((ext_vector_type(8))) float v8f;

#define BM 256
#define BN 128
#define BK 16
#define LDT 18  // padded LDS row stride (dwords): even (8B-aligned b64 reads) & conflict-free

// ---------------------------------------------------------------------------
// Stage 1: hidden = ReLU(relations[S,4] @ re_w1[4,D] + re_b1)   (K=4, VALU-only)
// ---------------------------------------------------------------------------
__global__ __launch_bounds__(256)
void relu_fc4_kernel(const float* __restrict__ rel4, const float* __restrict__ w1,
                     const float* __restrict__ b1, float* __restrict__ hidden,
                     int S, int D) {
  int idx = blockIdx.x * 256 + threadIdx.x;
  if (idx >= S * D) return;
  int s = idx / D, d = idx - s * D;
  float acc = b1[d];
#pragma unroll
  for (int k = 0; k < 4; ++k) acc += rel4[s * 4 + k] * w1[k * D + d];
  hidden[idx] = acc > 0.f ? acc : 0.f;
}

// ---------------------------------------------------------------------------
// Generic fp32 WMMA GEMM: C = A*B (+bias), batched over blockIdx.z.
//   A: row-major [M,K], lda; optional per-1024-column head gather:
//      if aHeadStride != 0: A(row,c) = Ab[(c/aKmod)*aHeadStride + row*lda + c%aKmod]
//   B: bRowMul==0 : NN, row-major [K,N], ldb  (transposed into LDS on load)
//      bRowMul>0  : NT with remap, B(k,n) = Bb[((c0+n)*bRowMul + z)*ldb + k]
//   256 threads = 8 waves (4x2); each wave 64x64 via 16x16x4 fp32 WMMA
//   (16 accumulator fragments -> 0.5 LDS fragment loads per WMMA).
// ---------------------------------------------------------------------------
__global__ __launch_bounds__(256)
void gemm_wmma_f32(const float* __restrict__ A, int lda, long aBatch,
                   long aHeadStride, int aKmod,
                   const float* __restrict__ B, int ldb, long bBatch, int bRowMul,
                   const float* __restrict__ bias,
                   float* __restrict__ C, int ldc, long cBatch,
                   int M, int N, int K) {
  __shared__ float As[2][BM * LDT];
  __shared__ float Bs[2][BN * LDT];

  const int tid = threadIdx.x;
  const int z = blockIdx.z;
  const float* __restrict__ Ab = A + (long)z * aBatch;
  const float* __restrict__ Bb = B + (long)z * bBatch;
  float* __restrict__ Cb = C + (long)z * cBatch;

  const int r0 = blockIdx.y * BM;
  const int c0 = blockIdx.x * BN;

  const int w = tid >> 5;
  const int lane = tid & 31;
  const int wm0 = (w & 3) * 64;        // 4 wave rows   (M)
  const int wn0 = (w >> 2) * 64;       // 2 wave cols   (N)
  const int fm = lane & 15;            // fragment row/col within 16
  const int kb = (lane >> 4) * 2;      // k-pair select (ISA 16x4 fp32 A layout)

  v8f acc[4][4];
#pragma unroll
  for (int i = 0; i < 4; ++i)
#pragma unroll
    for (int j = 0; j < 4; ++j)
#pragma unroll
      for (int r = 0; r < 8; ++r) acc[i][j][r] = 0.f;

  auto loadTiles = [&](int kt, int buf) {
    // --- A tile: BM x BK (256x16 = 1024 float4 across 256 threads) ---
#pragma unroll
    for (int j = 0; j < 4; ++j) {
      int e = tid + j * 256;
      int row = e >> 2;
      int col = (e & 3) * 4;
      const float* src;
      if (aHeadStride) {
        int c = kt + col;
        src = Ab + (long)(c / aKmod) * aHeadStride + (long)(r0 + row) * lda + (c % aKmod);
      } else {
        src = Ab + (long)(r0 + row) * lda + kt + col;
      }
      v4f v = *(const v4f*)src;
      float* dst = &As[buf][row * LDT + col];
      dst[0] = v.x; dst[1] = v.y; dst[2] = v.z; dst[3] = v.w;
    }
    // --- B tile -> LDS stored K-transposed: Bs[n][k] ---
    if (bRowMul == 0) {  // NN: read [k][n] rows, scatter-transpose
#pragma unroll
      for (int j = 0; j < 2; ++j) {
        int e = tid + j * 256;
        int k = e >> 5;
        int col = (e & 31) * 4;
        v4f v = *(const v4f*)(Bb + (long)(kt + k) * ldb + c0 + col);
        Bs[buf][(col + 0) * LDT + k] = v.x;
        Bs[buf][(col + 1) * LDT + k] = v.y;
        Bs[buf][(col + 2) * LDT + k] = v.z;
        Bs[buf][(col + 3) * LDT + k] = v.w;
      }
    } else {             // NT w/ row remap: rows already along K — direct copy
#pragma unroll
      for (int j = 0; j < 2; ++j) {
        int e = tid + j * 256;
        int n = e >> 2;
        int col = (e & 3) * 4;
        long row = (long)(c0 + n) * bRowMul + z;
        v4f v = *(const v4f*)(Bb + row * ldb + kt + col);
        float* dst = &Bs[buf][n * LDT + col];
        dst[0] = v.x; dst[1] = v.y; dst[2] = v.z; dst[3] = v.w;
      }
    }
  };

  loadTiles(0, 0);
  __syncthreads();
  const int nk = K / BK;
  for (int t = 0; t < nk; ++t) {
    int cur = t & 1;
    if (t + 1 < nk) loadTiles((t + 1) * BK, cur ^ 1);  // prefetch into other buffer
    const float* __restrict__ Al = &As[cur][0];
    const float* __restrict__ Bl = &Bs[cur][0];
#pragma unroll
    for (int kk = 0; kk < BK; kk += 4) {
      v2f af[4], bf[4];
#pragma unroll
      for (int mt = 0; mt < 4; ++mt)
        af[mt] = *(const v2f*)&Al[(wm0 + mt * 16 + fm) * LDT + kk + kb];
#pragma unroll
      for (int nt = 0; nt < 4; ++nt)
        bf[nt] = *(const v2f*)&Bl[(wn0 + nt * 16 + fm) * LDT + kk + kb];
#pragma unroll
      for (int mt = 0; mt < 4; ++mt)
#pragma unroll
        for (int nt = 0; nt < 4; ++nt)
          acc[mt][nt] = __builtin_amdgcn_wmma_f32_16x16x4_f32(
              false, af[mt], false, bf[nt], (short)0, acc[mt][nt], false, false);
    }
    __syncthreads();
  }

  // Epilogue: C/D layout — n = lane%16, m = r + 8*(lane>>4)
  const int fn = lane & 15;
  const int mhi = (lane >> 4) * 8;
#pragma unroll
  for (int mt = 0; mt < 4; ++mt)
#pragma unroll
    for (int nt = 0; nt < 4; ++nt) {
      int ncol = c0 + wn0 + nt * 16 + fn;
      float bv = bias ? bias[ncol] : 0.f;
#pragma unroll
      for (int r = 0; r < 8; ++r) {
        int mrow = r0 + wm0 + mt * 16 + mhi + r;
        Cb[(long)mrow * ldc + ncol] = acc[mt][nt][r] + bv;
      }
    }
}

// ---------------------------------------------------------------------------
// Row softmax, rowlen = 2048, 256 threads (8 elems/thread)
// ---------------------------------------------------------------------------
__global__ __launch_bounds__(256)
void softmax_kernel(float* __restrict__ data, int rowlen) {
  __shared__ float red[256];
  float* p = data + (long)blockIdx.x * rowlen;
  int tid = threadIdx.x;
  float v[8];
  float m = -1e30f;
#pragma unroll
  for (int i = 0; i < 8; ++i) { v[i] = p[tid + i * 256]; m = fmaxf(m, v[i]); }
  red[tid] = m; __syncthreads();
  for (int s = 128; s > 0; s >>= 1) {
    if (tid < s) red[tid] = fmaxf(red[tid], red[tid + s]);
    __syncthreads();
  }
  m = red[0]; __syncthreads();
  float sum = 0.f;
#pragma unroll
  for (int i = 0; i < 8; ++i) { v[i] = __expf(v[i] - m); sum += v[i]; }
  red[tid] = sum; __syncthreads();
  for (int s = 128; s > 0; s >>= 1) {
    if (tid < s) red[tid] += red[tid + s];
    __syncthreads();
  }
  float inv = 1.f / red[0];
#pragma unroll
  for (int i = 0; i < 8; ++i) p[tid + i * 256] = v[i] * inv;
}

// ---------------------------------------------------------------------------
extern "C" void kernel_launch(void* const* d_in, const int* in_sizes, int n_in,
                              void* d_out, int out_size, void* d_ws, size_t ws_size,
                              hipStream_t stream) {
  const int S = 2048, D = 1024, H = 8;
  const float* x         = (const float*)d_in[0];  // [S,D]
  const float* relations = (const float*)d_in[1];  // [S,4]
  const float* re_w1     = (const float*)d_in[2];  // [4,D]
  const float* re_b1     = (const float*)d_in[3];  // [D]
  const float* re_w2     = (const float*)d_in[4];  // [D,D]
  const float* re_b2     = (const float*)d_in[5];  // [D]
  const float* attn_w    = (const float*)d_in[6];  // [H,D,D]
  const float* out_w     = (const float*)d_in[7];  // [H*D,D]
  const float* out_b     = (const float*)d_in[8];  // [D]
  float* out = (float*)d_out;                      // [S,D]

  char* ws = (char*)d_ws;
  float* attnBuf   = (float*)ws;                              // [H,S,S]  128 MB
  float* scoresBuf = (float*)(ws + (size_t)H * S * S * 4);    // [H,S,D]   64 MB (reused for context)
  float* relBuf    = scoresBuf + (size_t)H * S * D;           // [S,D]      8 MB
  float* hidBuf    = relBuf + (size_t)S * D;                  // [S,D]      8 MB

  dim3 blk(256);

  // 1. hidden = ReLU(relations @ re_w1 + b1)
  relu_fc4_kernel<<<(S * D + 255) / 256, blk, 0, stream>>>(relations, re_w1, re_b1, hidBuf, S, D);

  // 2. rel = hidden @ re_w2 + b2                      [S,D]
  gemm_wmma_f32<<<dim3(D / BN, S / BM, 1), blk, 0, stream>>>(
      hidBuf, D, 0, 0, D, re_w2, D, 0, 0, re_b2, relBuf, D, 0, S, D, D);

  // 3. scores[h] = x @ attn_w[h]                      [H,S,D]
  gemm_wmma_f32<<<dim3(D / BN, S / BM, H), blk, 0, stream>>>(
      x, D, 0, 0, D, attn_w, D, (long)D * D, 0, nullptr, scoresBuf, D, (long)S * D, S, D, D);

  // 4. rel_scores[h][s,t] = rel[s,:] . scoresFlat[t*8+h,:]   (torch .view scramble via row remap)
  gemm_wmma_f32<<<dim3(S / BN, S / BM, H), blk, 0, stream>>>(
      relBuf, D, 0, 0, D, scoresBuf, D, 0, H, nullptr, attnBuf, S, (long)S * S, S, S, D);

  // 5. softmax rows (in place)
  softmax_kernel<<<H * S, blk, 0, stream>>>(attnBuf, S);

  // 6. context[h] = attn[h] @ x  -> overwrite scoresBuf
  gemm_wmma_f32<<<dim3(D / BN, S / BM, H), blk, 0, stream>>>(
      attnBuf, S, (long)S * S, 0, S, x, D, 0, 0, nullptr, scoresBuf, D, (long)S * D, S, D, S);

  // 7. out = contextPerm[S, H*D] @ out_w + out_b   (A head-gather: k -> context[k/1024][s][k%1024])
  gemm_wmma_f32<<<dim3(D / BN, S / BM, 1), blk, 0, stream>>>(
      scoresBuf, D, 0, (long)S * D, D, out_w, D, 0, 0, out_b, out, D, 0, S, D, H * D);
}